// Encoder_14224931684458
// MI455X (gfx1250) — compile-verified
//
#include <hip/hip_runtime.h>

#define B_  8
#define N_  4096
#define M_  256
#define KK_ 3
#define KN_ (KK_ * N_)

typedef float v2f __attribute__((ext_vector_type(2)));
typedef float v8f __attribute__((ext_vector_type(8)));

// ---------------------------------------------------------------------------
// FP32 WMMA 16x16x4:  D(16x16) = A(16x4) * B(4x16) + C
// A frag: a[u] = A[M = lane&15][K = u + 2*(lane>>4)]   (u=0,1 contiguous K!)
// B frag: b[u] = B[K = u + 2*(lane>>4)][N = lane&15]
// C/D   : c[u] = C[M = u + 8*(lane>>4)][N = lane&15]
// ---------------------------------------------------------------------------
__device__ __forceinline__ v8f wmma_k4(v2f a, v2f b, v8f c) {
  return __builtin_amdgcn_wmma_f32_16x16x4_f32(false, a, false, b, (short)0, c,
                                               false, false);
}

// One 16x16 output tile of OUT = Wp * act^T, K-padded so the loop is branch
// free: Wp is zero-padded row-major (Cout x wStride), act is point-major LDS
// [point][channel] with row stride actStride (strides chosen == 4 mod 64 so
// half-wave B-loads are bank-conflict free). Both sides load 8B per step.
__device__ __forceinline__ v8f mma_tile_t(const float* __restrict__ Wp, int wStride,
                                          const float* __restrict__ act, int actStride,
                                          int kLoop, int cout0, int p0, int lane) {
  const int col = lane & 15;
  const int hi  = lane >> 4;
  v8f acc = {};
  const float* wrow = Wp + (size_t)(cout0 + col) * (size_t)wStride + 2 * hi;
  const float* arow = act + (size_t)(p0 + col) * (size_t)actStride + 2 * hi;
#pragma unroll 2
  for (int k0 = 0; k0 < kLoop; k0 += 4) {
    const v2f a = *(const v2f*)(wrow + k0);   // global_load_b64 (L2-resident W)
    const v2f b = *(const v2f*)(arow + k0);   // ds_load_b64
    acc = wmma_k4(a, b, acc);
  }
  return acc;
}

// Bias + ReLU + point-major store: each lane writes 8 contiguous floats.
__device__ __forceinline__ void store_relu_t(float* __restrict__ out, int outStride,
                                             v8f acc, const float* __restrict__ bias,
                                             int cout0, int p0, int lane) {
  const int col = lane & 15;
  const int hi  = lane >> 4;
  float* dst = out + (size_t)(p0 + col) * (size_t)outStride + cout0 + 8 * hi;
  const float* bs = bias + cout0 + 8 * hi;
#pragma unroll
  for (int u = 0; u < 8; ++u)
    dst[u] = fmaxf(acc[u] + bs[u], 0.0f);
}

// ---------------------------------------------------------------------------
// Zero-pad odd-K weights once per launch (3->4, 259->260, 387->388, 1155->1156)
// ---------------------------------------------------------------------------
__global__ void pad_w_kernel(const float* __restrict__ src, float* __restrict__ dst,
                             int Cout, int Cin, int CinPad) {
  const int i = blockIdx.x * blockDim.x + threadIdx.x;
  if (i >= Cout * CinPad) return;
  const int r = i / CinPad, c = i % CinPad;
  dst[i] = (c < Cin) ? src[r * Cin + c] : 0.0f;
}

__global__ void zero_kernel(float* __restrict__ p, int n) {
  const int i = blockIdx.x * blockDim.x + threadIdx.x;
  if (i < n) p[i] = 0.0f;
}

// ---------------------------------------------------------------------------
// 1) 3-NN of each point vs 256 SOM nodes (nodes staged in LDS)
// ---------------------------------------------------------------------------
__global__ __launch_bounds__(256) void knn_kernel(const float* __restrict__ x,
                                                  const float* __restrict__ node,
                                                  int* __restrict__ minIdx) {
  const int blocksPerB = N_ / 256;
  const int b  = blockIdx.x / blocksPerB;
  const int n0 = (blockIdx.x % blocksPerB) * 256;
  __shared__ float sn[3 * M_];
  for (int i = threadIdx.x; i < 3 * M_; i += 256)
    sn[i] = node[(size_t)b * 3 * M_ + i];
  __syncthreads();
  const int n = n0 + threadIdx.x;
  const float px = x[((size_t)b * 3 + 0) * N_ + n];
  const float py = x[((size_t)b * 3 + 1) * N_ + n];
  const float pz = x[((size_t)b * 3 + 2) * N_ + n];
  float d0 = 3.4e38f, d1 = 3.4e38f, d2 = 3.4e38f;
  int   i0 = 0, i1 = 0, i2 = 0;
  for (int m = 0; m < M_; ++m) {
    const float dx = px - sn[0 * M_ + m];
    const float dy = py - sn[1 * M_ + m];
    const float dz = pz - sn[2 * M_ + m];
    const float d  = dx * dx + dy * dy + dz * dz;
    if (d < d0)      { d2 = d1; i2 = i1; d1 = d0; i1 = i0; d0 = d; i0 = m; }
    else if (d < d1) { d2 = d1; i2 = i1; d1 = d;  i1 = m; }
    else if (d < d2) { d2 = d;  i2 = m; }
  }
  minIdx[(size_t)b * KN_ + 0 * N_ + n] = i0;
  minIdx[(size_t)b * KN_ + 1 * N_ + n] = i1;
  minIdx[(size_t)b * KN_ + 2 * N_ + n] = i2;
}

// ---------------------------------------------------------------------------
// 2) Deterministic per-node segment count + mean (one block per (b, node))
// ---------------------------------------------------------------------------
__global__ __launch_bounds__(256) void seg_mean_kernel(
    const float* __restrict__ x, const int* __restrict__ minIdx,
    float* __restrict__ mean, float* __restrict__ counts) {
  const int b = blockIdx.x / M_;
  const int m = blockIdx.x % M_;
  const int tid = threadIdx.x;
  float cnt = 0.f, sx = 0.f, sy = 0.f, sz = 0.f;
  for (int j = tid; j < KN_; j += 256) {
    if (minIdx[(size_t)b * KN_ + j] == m) {
      const int n = j % N_;
      cnt += 1.0f;
      sx += x[((size_t)b * 3 + 0) * N_ + n];
      sy += x[((size_t)b * 3 + 1) * N_ + n];
      sz += x[((size_t)b * 3 + 2) * N_ + n];
    }
  }
  __shared__ float red[4][256];
  red[0][tid] = cnt; red[1][tid] = sx; red[2][tid] = sy; red[3][tid] = sz;
  __syncthreads();
  for (int s = 128; s > 0; s >>= 1) {
    if (tid < s) {
#pragma unroll
      for (int c = 0; c < 4; ++c) red[c][tid] += red[c][tid + s];
    }
    __syncthreads();
  }
  if (tid == 0) {
    const float c0 = red[0][0];
    counts[(size_t)b * M_ + m] = c0;
    const float inv = 1.0f / (c0 + 1e-5f);
    mean[((size_t)b * 3 + 0) * M_ + m] = red[1][0] * inv;
    mean[((size_t)b * 3 + 1) * M_ + m] = red[2][0] * inv;
    mean[((size_t)b * 3 + 2) * M_ + m] = red[3][0] * inv;
  }
}

// ---------------------------------------------------------------------------
// 3) Fused point-resnet #1 over a 32-point tile, all layers in LDS.
//    Point-major buffers; x_dec embedded at columns 256..259 of h2x so the
//    layer-3 concat is one contiguous 260-wide K range.  Strides 68/132/260
//    are == 4 mod 64 -> bank-conflict-free B loads.  Layer-3 folds straight
//    into per-node atomic segment-max (float-bits max, valid for ReLU >= 0).
// ---------------------------------------------------------------------------
__global__ __launch_bounds__(256) void first_resnet_kernel(
    const float* __restrict__ x, const int* __restrict__ minIdx,
    const float* __restrict__ mean,
    const float* __restrict__ w0p, const float* __restrict__ b0,  // 64x4 padded
    const float* __restrict__ w1,  const float* __restrict__ b1,  // 128x64
    const float* __restrict__ w2,  const float* __restrict__ b2,  // 256x128
    const float* __restrict__ w3p, const float* __restrict__ b3,  // 384x260 padded
    int* __restrict__ nodeMax, float* __restrict__ fallback) {
  const int tilesPerB = KN_ / 32;
  const int b    = blockIdx.x / tilesPerB;
  const int tile = blockIdx.x % tilesPerB;
  const int j0   = tile * 32;

  __shared__ __align__(16) float h2x[32 * 260];  // [pt][ch]: 0..255 h2, 256..258 xd, 259 = 0
  __shared__ __align__(16) float h0[32 * 68];    // cols 0..63
  __shared__ __align__(16) float h1[32 * 132];   // cols 0..127
  __shared__ int midx[32];

  const int tid = threadIdx.x;
  if (tid < 32) {
    const int j = j0 + tid;
    const int m = minIdx[(size_t)b * KN_ + j];
    midx[tid] = m;
    const int n = j % N_;
#pragma unroll
    for (int c = 0; c < 3; ++c)
      h2x[tid * 260 + 256 + c] =
          x[((size_t)b * 3 + c) * N_ + n] - mean[((size_t)b * 3 + c) * M_ + m];
    h2x[tid * 260 + 259] = 0.0f;
  }
  __syncthreads();

  const int wave = tid >> 5;
  const int lane = tid & 31;

  // L0: 64 <- 4 (padded x_dec)   4 cout tiles x 2 point tiles
  for (int t = wave; t < 8; t += 8) {
    const int ct = t >> 1, pt = t & 1;
    v8f acc = mma_tile_t(w0p, 4, h2x + 256, 260, 4, ct * 16, pt * 16, lane);
    store_relu_t(h0, 68, acc, b0, ct * 16, pt * 16, lane);
  }
  __syncthreads();
  // L1: 128 <- 64
  for (int t = wave; t < 16; t += 8) {
    const int ct = t >> 1, pt = t & 1;
    v8f acc = mma_tile_t(w1, 64, h0, 68, 64, ct * 16, pt * 16, lane);
    store_relu_t(h1, 132, acc, b1, ct * 16, pt * 16, lane);
  }
  __syncthreads();
  // L2: 256 <- 128  (writes h2x cols 0..255; xd cols untouched)
  for (int t = wave; t < 32; t += 8) {
    const int ct = t >> 1, pt = t & 1;
    v8f acc = mma_tile_t(w2, 128, h1, 132, 128, ct * 16, pt * 16, lane);
    store_relu_t(h2x, 260, acc, b2, ct * 16, pt * 16, lane);
  }
  __syncthreads();
  // L3: 384 <- concat(256, x_dec)  -> atomic segment max per node
  for (int t = wave; t < 48; t += 8) {
    const int ct = t >> 1, pt = t & 1;
    v8f acc = mma_tile_t(w3p, 260, h2x, 260, 260, ct * 16, pt * 16, lane);
    const int col = lane & 15, hi = lane >> 4;
    const int p = pt * 16 + col;
    const int m = midx[p];
#pragma unroll
    for (int u = 0; u < 8; ++u) {
      const int row = ct * 16 + u + 8 * hi;
      const float v = fmaxf(acc[u] + b3[row], 0.0f);
      atomicMax(&nodeMax[((size_t)b * 384 + row) * M_ + m], __float_as_int(v));
      if (j0 + p == 0) fallback[(size_t)b * 384 + row] = v;  // first[:, :, 0]
    }
  }
}

// ---------------------------------------------------------------------------
// 4) Fused node-resnet #2 over a 16-node tile.  final_in (388 incl. zero pad)
//    is built directly into bufA columns 768..1155, so layer 0 reads it there
//    and layer 3 reads one contiguous 1156-wide concat(h2, final_in).
//    Column ranges are disjoint within each layer -> race-free ping-pong.
// ---------------------------------------------------------------------------
#define SRN_SA 1156
#define SRN_SB 516
__global__ __launch_bounds__(256) void second_resnet_kernel(
    const float* __restrict__ mean, const float* __restrict__ counts,
    const int* __restrict__ nodeMax, const float* __restrict__ fallback,
    const float* __restrict__ w0p, const float* __restrict__ b0,  // 512x388 padded
    const float* __restrict__ w1,  const float* __restrict__ b1,  // 512x512
    const float* __restrict__ w2,  const float* __restrict__ b2,  // 768x512
    const float* __restrict__ w3p, const float* __restrict__ b3,  // 1024x1156 padded
    int* __restrict__ outBits) {
  extern __shared__ float smem[];
  float* bufA = smem;                       // 16 x 1156: 0..767 acts, 768..1155 final_in
  float* bufB = bufA + 16 * SRN_SA;         // 16 x 516

  const int b  = blockIdx.x / (M_ / 16);
  const int m0 = (blockIdx.x % (M_ / 16)) * 16;
  const int tid = threadIdx.x;

  for (int idx = tid; idx < 16 * 388; idx += 256) {
    const int p = idx / 388;
    const int r = idx % 388;
    const int m = m0 + p;
    float v = 0.0f;
    if (r < 3) {
      v = mean[((size_t)b * 3 + r) * M_ + m];
    } else if (r < 387) {
      const int c = r - 3;
      const bool has = counts[(size_t)b * M_ + m] > 0.0f;
      v = has ? __int_as_float(nodeMax[((size_t)b * 384 + c) * M_ + m])
              : fallback[(size_t)b * 384 + c];
    }
    bufA[p * SRN_SA + 768 + r] = v;
  }
  __syncthreads();

  const int wave = tid >> 5;
  const int lane = tid & 31;

  // L0: 512 <- 388 (reads cols 768.., writes cols 0..511: disjoint)
  for (int t = wave; t < 32; t += 8) {
    v8f acc = mma_tile_t(w0p, 388, bufA + 768, SRN_SA, 388, t * 16, 0, lane);
    store_relu_t(bufA, SRN_SA, acc, b0, t * 16, 0, lane);
  }
  __syncthreads();
  // L1: 512 <- 512   bufA[0..511] -> bufB
  for (int t = wave; t < 32; t += 8) {
    v8f acc = mma_tile_t(w1, 512, bufA, SRN_SA, 512, t * 16, 0, lane);
    store_relu_t(bufB, SRN_SB, acc, b1, t * 16, 0, lane);
  }
  __syncthreads();
  // L2: 768 <- 512   bufB -> bufA cols 0..767
  for (int t = wave; t < 48; t += 8) {
    v8f acc = mma_tile_t(w2, 512, bufB, SRN_SB, 512, t * 16, 0, lane);
    store_relu_t(bufA, SRN_SA, acc, b2, t * 16, 0, lane);
  }
  __syncthreads();
  // L3: 1024 <- contiguous concat(768 acts, 388 final_in) -> global max
  for (int t = wave; t < 64; t += 8) {
    v8f acc = mma_tile_t(w3p, SRN_SA, bufA, SRN_SA, SRN_SA, t * 16, 0, lane);
    const int hi = lane >> 4;
#pragma unroll
    for (int u = 0; u < 8; ++u) {
      const int row = t * 16 + u + 8 * hi;
      const float v = fmaxf(acc[u] + b3[row], 0.0f);
      atomicMax(&outBits[(size_t)b * 1024 + row], __float_as_int(v));
    }
  }
}

// ---------------------------------------------------------------------------
extern "C" void kernel_launch(void* const* d_in, const int* in_sizes, int n_in,
                              void* d_out, int out_size, void* d_ws,
                              size_t ws_size, hipStream_t stream) {
  const float* x    = (const float*)d_in[0];
  const float* node = (const float*)d_in[2];
  const float* fpw0 = (const float*)d_in[4];
  const float* fpb0 = (const float*)d_in[5];
  const float* fpw1 = (const float*)d_in[6];
  const float* fpb1 = (const float*)d_in[7];
  const float* fpw2 = (const float*)d_in[8];
  const float* fpb2 = (const float*)d_in[9];
  const float* fpw3 = (const float*)d_in[10];
  const float* fpb3 = (const float*)d_in[11];
  const float* fnw0 = (const float*)d_in[12];
  const float* fnb0 = (const float*)d_in[13];
  const float* fnw1 = (const float*)d_in[14];
  const float* fnb1 = (const float*)d_in[15];
  const float* fnw2 = (const float*)d_in[16];
  const float* fnb2 = (const float*)d_in[17];
  const float* fnw3 = (const float*)d_in[18];
  const float* fnb3 = (const float*)d_in[19];

  char* ws = (char*)d_ws;
  int*   minIdx   = (int*)ws;    ws += (size_t)B_ * KN_ * 4;
  float* counts   = (float*)ws;  ws += (size_t)B_ * M_ * 4;
  float* meanBuf  = (float*)ws;  ws += (size_t)B_ * 3 * M_ * 4;
  int*   nodeMax  = (int*)ws;    ws += (size_t)B_ * 384 * M_ * 4;
  float* fallback = (float*)ws;  ws += (size_t)B_ * 384 * 4;
  float* wp_fp0   = (float*)ws;  ws += (size_t)64 * 4 * 4;
  float* wp_fp3   = (float*)ws;  ws += (size_t)384 * 260 * 4;
  float* wp_fn0   = (float*)ws;  ws += (size_t)512 * 388 * 4;
  float* wp_fn3   = (float*)ws;  ws += (size_t)1024 * 1156 * 4;

  // Zero-pad the odd-K weight matrices (branch-free WMMA K loops).
  pad_w_kernel<<<(64 * 4 + 255) / 256, 256, 0, stream>>>(fpw0, wp_fp0, 64, 3, 4);
  pad_w_kernel<<<(384 * 260 + 255) / 256, 256, 0, stream>>>(fpw3, wp_fp3, 384, 259, 260);
  pad_w_kernel<<<(512 * 388 + 255) / 256, 256, 0, stream>>>(fnw0, wp_fn0, 512, 387, 388);
  pad_w_kernel<<<(1024 * 1156 + 255) / 256, 256, 0, stream>>>(fnw3, wp_fn3, 1024, 1155, 1156);

  // Zero-init atomic-max targets (float 0.0 == int 0; all maxed values >= 0).
  {
    const int nNM = B_ * 384 * M_;
    zero_kernel<<<(nNM + 255) / 256, 256, 0, stream>>>((float*)nodeMax, nNM);
    zero_kernel<<<(out_size + 255) / 256, 256, 0, stream>>>((float*)d_out, out_size);
  }

  knn_kernel<<<B_ * (N_ / 256), 256, 0, stream>>>(x, node, minIdx);
  seg_mean_kernel<<<B_ * M_, 256, 0, stream>>>(x, minIdx, meanBuf, counts);

  first_resnet_kernel<<<B_ * (KN_ / 32), 256, 0, stream>>>(
      x, minIdx, meanBuf, wp_fp0, fpb0, fpw1, fpb1, fpw2, fpb2, wp_fp3, fpb3,
      nodeMax, fallback);

  const int smem2 = (SRN_SA + SRN_SB) * 16 * (int)sizeof(float);  // ~104.5 KB
  (void)hipFuncSetAttribute((const void*)second_resnet_kernel,
                            hipFuncAttributeMaxDynamicSharedMemorySize, smem2);
  second_resnet_kernel<<<B_ * (M_ / 16), 256, smem2, stream>>>(
      meanBuf, counts, nodeMax, fallback, wp_fn0, fnb0, fnw1, fnb1, fnw2, fnb2,
      wp_fn3, fnb3, (int*)d_out);
}